// GAT_48473000902934
// MI455X (gfx1250) — compile-verified
//
#include <hip/hip_runtime.h>
#include <hip/hip_bf16.h>

typedef float v2f __attribute__((ext_vector_type(2)));
typedef float v8f __attribute__((ext_vector_type(8)));

#define F_IN   512
#define HEADS  8
#define HID    8
#define HO1    64        // HEADS*HID
#define CLS    16
#define LRELU  0.2f

// ---- monotonic float<->uint encoding for atomicMax-based segment max ----
__device__ __forceinline__ unsigned fenc(float f) {
    unsigned b = __float_as_uint(f);
    return (b & 0x80000000u) ? ~b : (b | 0x80000000u);
}
__device__ __forceinline__ float fdec(unsigned u) {
    return __uint_as_float((u & 0x80000000u) ? (u ^ 0x80000000u) : ~u);
}

__device__ __forceinline__ void edge_sd(int e, int E, const long long* es,
                                        const long long* ed, int& src, int& dst) {
    if (e < E) { src = (int)es[e]; dst = (int)ed[e]; }
    else       { src = e - E; dst = e - E; }   // appended self loops
}

// ---------------- init all accumulators / outputs ----------------
__global__ void k_init(float* out1, unsigned* m1, float* den1,
                       unsigned* m2, float* den2, float* outf, int n) {
    int i = blockIdx.x * 256 + threadIdx.x;
    if (i < n * HO1)   out1[i] = 0.f;
    if (i < n * HEADS) { m1[i] = 0u; den1[i] = 0.f; }
    if (i < n * CLS)   outf[i] = 0.f;
    if (i < n)         { m2[i] = 0u; den2[i] = 0.f; }
}

// ---------------- GEMM1: xl1[N,64] = x[N,512] @ W1[512,64] (fp32 WMMA) ----------------
// W1 chunk is stored K-pair-interleaved in LDS: element (k,n) -> sW[(k>>1)*128 + 2n + (k&1)]
// so a B fragment {W[kk][n], W[kk+1][n]} is a single aligned ds_load_b64.
__global__ __launch_bounds__(256) void k_gemm1(const float* __restrict__ x,
                                               const float* __restrict__ W1,
                                               float* __restrict__ xl1, int n) {
    __shared__ float sW[128 * HO1];           // 32 KB K-chunk of W1, interleaved
    const int tid  = threadIdx.x;
    const int wave = tid >> 5, lane = tid & 31;
    const int m    = lane & 15, half = lane >> 4;
    const int row0 = (blockIdx.x * 8 + wave) * 16;
    int r  = row0 + m;
    int rc = r < n ? r : n - 1;               // clamp (no branch -> EXEC stays full)
    const float* xrow = x + (size_t)rc * F_IN;

    v8f acc[4] = {};
    for (int kc = 0; kc < F_IN; kc += 128) {
        __syncthreads();
        for (int i = tid; i < 128 * HO1; i += 256) {
            int k = i >> 6, nn = i & 63;      // within-chunk row k, col nn
            sW[(k >> 1) * 128 + (nn << 1) + (k & 1)] = W1[kc * HO1 + i];
        }
        __syncthreads();
        for (int k = 0; k < 128; k += 4) {
            const int p = (k >> 1) + half;    // K-pair index for this half-wave
            v2f A = *(const v2f*)(xrow + kc + k + 2 * half);  // A: M=lane%16, K pair
            const float* bp = sW + p * 128 + (m << 1);
#pragma unroll
            for (int t = 0; t < 4; ++t) {
                v2f B = *(const v2f*)(bp + (t << 5));          // one ds_load_b64
                acc[t] = __builtin_amdgcn_wmma_f32_16x16x4_f32(
                    false, A, false, B, (short)0, acc[t], false, false);
            }
        }
    }
#pragma unroll
    for (int j = 0; j < 8; ++j) {             // D: VGPR j -> row j (+8 for hi half)
        int row = row0 + j + 8 * half;
        if (row < n) {
#pragma unroll
            for (int t = 0; t < 4; ++t)
                xl1[(size_t)row * HO1 + t * 16 + m] = acc[t][j];
        }
    }
}

// ---------------- per-node attention scores, layer 1 ----------------
__global__ void k_scores1(const float* __restrict__ xl1, const float* __restrict__ as,
                          const float* __restrict__ ad, float* s1, float* d1, int n) {
    int i = blockIdx.x * 256 + threadIdx.x;
    if (i >= n * HEADS) return;
    int nn = i >> 3, h = i & 7;
    const float* v = xl1 + (size_t)nn * HO1 + h * HID;
    float ss = 0.f, dd = 0.f;
#pragma unroll
    for (int c = 0; c < HID; ++c) { ss += v[c] * as[h * HID + c]; dd += v[c] * ad[h * HID + c]; }
    s1[i] = ss; d1[i] = dd;
}

// ---------------- layer-1 edge passes ----------------
__global__ void k_emax1(const long long* es, const long long* ed,
                        const float* __restrict__ s1, const float* __restrict__ d1,
                        unsigned* m1, int E, int ET) {
    int i = blockIdx.x * 256 + threadIdx.x;
    if (i >= ET * HEADS) return;
    int e = i >> 3, h = i & 7, src, dst;
    edge_sd(e, E, es, ed, src, dst);
    float a = s1[src * HEADS + h] + d1[dst * HEADS + h];
    a = a > 0.f ? a : LRELU * a;
    atomicMax(&m1[dst * HEADS + h], fenc(a));
}

__global__ void k_esum1(const long long* es, const long long* ed,
                        const float* __restrict__ s1, const float* __restrict__ d1,
                        const unsigned* __restrict__ m1, float* den1, int E, int ET) {
    int i = blockIdx.x * 256 + threadIdx.x;
    if (i >= ET * HEADS) return;
    int e = i >> 3, h = i & 7, src, dst;
    edge_sd(e, E, es, ed, src, dst);
    float a = s1[src * HEADS + h] + d1[dst * HEADS + h];
    a = a > 0.f ? a : LRELU * a;
    atomicAdd(&den1[dst * HEADS + h], expf(a - fdec(m1[dst * HEADS + h])));
}

// one wave per edge; each lane handles 2 of the 64 channels
__global__ void k_agg1(const long long* es, const long long* ed,
                       const float* __restrict__ s1, const float* __restrict__ d1,
                       const unsigned* __restrict__ m1, const float* __restrict__ den1,
                       const float* __restrict__ xl1, float* out1, int E, int ET) {
    int e = blockIdx.x * 8 + (threadIdx.x >> 5);
    if (e >= ET) return;
    int lane = threadIdx.x & 31, src, dst;
    edge_sd(e, E, es, ed, src, dst);
    int h = lane >> 2;                         // channels 2*lane,2*lane+1 -> head
    float a = s1[src * HEADS + h] + d1[dst * HEADS + h];
    a = a > 0.f ? a : LRELU * a;
    float w = expf(a - fdec(m1[dst * HEADS + h])) / (den1[dst * HEADS + h] + 1e-16f);
    v2f xv = *(const v2f*)(xl1 + (size_t)src * HO1 + 2 * lane);
    atomicAdd(&out1[(size_t)dst * HO1 + 2 * lane],     w * xv.x);
    atomicAdd(&out1[(size_t)dst * HO1 + 2 * lane + 1], w * xv.y);
}

// ---------------- bias + ELU (in place) ----------------
__global__ void k_elu(float* out1, const float* __restrict__ b1, int n) {
    int i = blockIdx.x * 256 + threadIdx.x;
    if (i >= n * HO1) return;
    float v = out1[i] + b1[i % HO1];
    out1[i] = v > 0.f ? v : expm1f(v);
}

// ---------------- GEMM2: xl2[N,16] = h[N,64] @ W2[64,16] (fp32 WMMA) ----------------
__global__ __launch_bounds__(256) void k_gemm2(const float* __restrict__ h,
                                               const float* __restrict__ W2,
                                               float* __restrict__ xl2, int n) {
    __shared__ float sW[HO1 * CLS];           // 4 KB, K-pair interleaved
    const int tid = threadIdx.x;
    for (int i = tid; i < HO1 * CLS; i += 256) {
        int k = i >> 4, nn = i & 15;
        sW[(k >> 1) * 32 + (nn << 1) + (k & 1)] = W2[i];
    }
    __syncthreads();
    const int wave = tid >> 5, lane = tid & 31;
    const int m = lane & 15, half = lane >> 4;
    const int row0 = (blockIdx.x * 8 + wave) * 16;
    int r = row0 + m;
    int rc = r < n ? r : n - 1;
    const float* hr = h + (size_t)rc * HO1;
    v8f acc = {};
    for (int k = 0; k < HO1; k += 4) {
        const int p = (k >> 1) + half;
        v2f A = *(const v2f*)(hr + k + 2 * half);
        v2f B = *(const v2f*)(sW + p * 32 + (m << 1));
        acc = __builtin_amdgcn_wmma_f32_16x16x4_f32(
            false, A, false, B, (short)0, acc, false, false);
    }
#pragma unroll
    for (int j = 0; j < 8; ++j) {
        int row = row0 + j + 8 * half;
        if (row < n) xl2[(size_t)row * CLS + m] = acc[j];
    }
}

// ---------------- per-node scores, layer 2 (H=1, C=16) ----------------
__global__ void k_scores2(const float* __restrict__ xl2, const float* __restrict__ as,
                          const float* __restrict__ ad, float* s2, float* d2, int n) {
    int i = blockIdx.x * 256 + threadIdx.x;
    if (i >= n) return;
    const float* v = xl2 + (size_t)i * CLS;
    float ss = 0.f, dd = 0.f;
#pragma unroll
    for (int c = 0; c < CLS; ++c) { ss += v[c] * as[c]; dd += v[c] * ad[c]; }
    s2[i] = ss; d2[i] = dd;
}

__global__ void k_emax2(const long long* es, const long long* ed,
                        const float* __restrict__ s2, const float* __restrict__ d2,
                        unsigned* m2, int E, int ET) {
    int e = blockIdx.x * 256 + threadIdx.x;
    if (e >= ET) return;
    int src, dst; edge_sd(e, E, es, ed, src, dst);
    float a = s2[src] + d2[dst];
    a = a > 0.f ? a : LRELU * a;
    atomicMax(&m2[dst], fenc(a));
}

__global__ void k_esum2(const long long* es, const long long* ed,
                        const float* __restrict__ s2, const float* __restrict__ d2,
                        const unsigned* __restrict__ m2, float* den2, int E, int ET) {
    int e = blockIdx.x * 256 + threadIdx.x;
    if (e >= ET) return;
    int src, dst; edge_sd(e, E, es, ed, src, dst);
    float a = s2[src] + d2[dst];
    a = a > 0.f ? a : LRELU * a;
    atomicAdd(&den2[dst], expf(a - fdec(m2[dst])));
}

__global__ void k_agg2(const long long* es, const long long* ed,
                       const float* __restrict__ s2, const float* __restrict__ d2,
                       const unsigned* __restrict__ m2, const float* __restrict__ den2,
                       const float* __restrict__ xl2, float* outf, int E, int ET) {
    int i = blockIdx.x * 256 + threadIdx.x;
    if (i >= ET * CLS) return;
    int e = i >> 4, c = i & 15, src, dst;
    edge_sd(e, E, es, ed, src, dst);
    float a = s2[src] + d2[dst];
    a = a > 0.f ? a : LRELU * a;
    float w = expf(a - fdec(m2[dst])) / (den2[dst] + 1e-16f);
    atomicAdd(&outf[(size_t)dst * CLS + c], w * xl2[(size_t)src * CLS + c]);
}

// ---------------- bias + log_softmax (in place on d_out) ----------------
__global__ void k_lsm(float* outf, const float* __restrict__ b2, int n) {
    int i = blockIdx.x * 256 + threadIdx.x;
    if (i >= n) return;
    float v[CLS];
    float mx = -3.402823466e+38f;
#pragma unroll
    for (int c = 0; c < CLS; ++c) { v[c] = outf[(size_t)i * CLS + c] + b2[c]; mx = fmaxf(mx, v[c]); }
    float s = 0.f;
#pragma unroll
    for (int c = 0; c < CLS; ++c) s += expf(v[c] - mx);
    float lse = mx + logf(s);
#pragma unroll
    for (int c = 0; c < CLS; ++c) outf[(size_t)i * CLS + c] = v[c] - lse;
}

extern "C" void kernel_launch(void* const* d_in, const int* in_sizes, int n_in,
                              void* d_out, int out_size, void* d_ws, size_t ws_size,
                              hipStream_t stream) {
    const float*     x    = (const float*)d_in[0];
    const long long* ei   = (const long long*)d_in[1];
    const float*     W1   = (const float*)d_in[2];
    const float*     as1  = (const float*)d_in[3];
    const float*     ad1  = (const float*)d_in[4];
    const float*     b1   = (const float*)d_in[5];
    const float*     W2   = (const float*)d_in[6];
    const float*     as2  = (const float*)d_in[7];
    const float*     ad2  = (const float*)d_in[8];
    const float*     b2   = (const float*)d_in[9];
    float*           outf = (float*)d_out;

    const int N  = in_sizes[0] / F_IN;        // 100000
    const int E  = in_sizes[1] / 2;           // 1600000
    const int ET = E + N;                     // with self loops
    const long long* es = ei;
    const long long* ed = ei + E;

    // workspace carve-up (~72 MB)
    float* ws = (float*)d_ws;
    float*    xl1  = ws;               ws += (size_t)N * HO1;
    float*    out1 = ws;               ws += (size_t)N * HO1;
    float*    s1   = ws;               ws += (size_t)N * HEADS;
    float*    d1   = ws;               ws += (size_t)N * HEADS;
    unsigned* m1   = (unsigned*)ws;    ws += (size_t)N * HEADS;
    float*    den1 = ws;               ws += (size_t)N * HEADS;
    float*    xl2  = ws;               ws += (size_t)N * CLS;
    float*    s2   = ws;               ws += (size_t)N;
    float*    d2   = ws;               ws += (size_t)N;
    unsigned* m2   = (unsigned*)ws;    ws += (size_t)N;
    float*    den2 = ws;               ws += (size_t)N;

    const int B = 256;
    auto cdiv = [](long long a, long long b) { return (int)((a + b - 1) / b); };

    k_init<<<cdiv((long long)N * HO1, B), B, 0, stream>>>(out1, m1, den1, m2, den2, outf, N);
    k_gemm1<<<cdiv(N, 128), B, 0, stream>>>(x, W1, xl1, N);
    k_scores1<<<cdiv((long long)N * HEADS, B), B, 0, stream>>>(xl1, as1, ad1, s1, d1, N);
    k_emax1<<<cdiv((long long)ET * HEADS, B), B, 0, stream>>>(es, ed, s1, d1, m1, E, ET);
    k_esum1<<<cdiv((long long)ET * HEADS, B), B, 0, stream>>>(es, ed, s1, d1, m1, den1, E, ET);
    k_agg1<<<cdiv(ET, 8), B, 0, stream>>>(es, ed, s1, d1, m1, den1, xl1, out1, E, ET);
    k_elu<<<cdiv((long long)N * HO1, B), B, 0, stream>>>(out1, b1, N);
    k_gemm2<<<cdiv(N, 128), B, 0, stream>>>(out1, W2, xl2, N);
    k_scores2<<<cdiv(N, B), B, 0, stream>>>(xl2, as2, ad2, s2, d2, N);
    k_emax2<<<cdiv(ET, B), B, 0, stream>>>(es, ed, s2, d2, m2, E, ET);
    k_esum2<<<cdiv(ET, B), B, 0, stream>>>(es, ed, s2, d2, m2, den2, E, ET);
    k_agg2<<<cdiv((long long)ET * CLS, B), B, 0, stream>>>(es, ed, s2, d2, m2, den2, xl2, outf, E, ET);
    k_lsm<<<cdiv(N, B), B, 0, stream>>>(outf, b2, N);
}